// SVF_65635690217762
// MI455X (gfx1250) — compile-verified
//
#include <hip/hip_runtime.h>

// SVF scaling-and-squaring integration, 160^3 x 3 channels, f32.
// Memory-bound gather workload: working set (in+out+grid = 147 MB) fits the
// MI455X 192 MB L2, so we ping-pong buffers and all 12 iterations run at L2
// bandwidth. No matrix contraction exists -> no WMMA by design.
// Key optimization this round: x-adjacent trilinear corners fetched as one
// B64 pair load (halves gather request count), with boundary handling folded
// into zero weights (exactly matches grid_sample padding_mode='zeros').

constexpr int Dd = 160, Hh = 160, Ww = 160;
constexpr int NVOX = Dd * Hh * Ww;          // 4,096,000 voxels
constexpr int NO_STEPS = 12;

// 4-byte-aligned float pair: lets the compiler emit global_load_b64 at DWORD
// alignment (legal on gfx1250: DWORD mode ignores 2 LSBs for DWORD+ ops).
struct __attribute__((packed, aligned(4))) fpair { float x, y; };

// ---------------------------------------------------------------- init ------
// out[i] = v[i] * 2^-12, vectorized B128. 3*NVOX divisible by 4.
__global__ __launch_bounds__(256)
void svf_scale_kernel(const float* __restrict__ v, float* __restrict__ out) {
    const int i = blockIdx.x * blockDim.x + threadIdx.x;
    const float s = 1.0f / 4096.0f;
    const float4 a = reinterpret_cast<const float4*>(v)[i];
    reinterpret_cast<float4*>(out)[i] = make_float4(a.x * s, a.y * s, a.z * s, a.w * s);
}

// ---------------------------------------------------------------- iter ------
// One scaling-and-squaring step:
//   out[c, p] = in[c, p] + trilinear(in[c], X(p)), with
//   X = (((grid + in_transposed) + 1) * dim - 1) * 0.5   (align_corners=False)
// Each thread handles 4 consecutive x-voxels so every streaming access is a
// 128-bit vector op; gathers are 64-bit x-pair loads.
__global__ __launch_bounds__(256)
void svf_iter_kernel(const float* __restrict__ in,
                     const float* __restrict__ grid,
                     float* __restrict__ out) {
    const int t  = blockIdx.x * blockDim.x + threadIdx.x;  // 0 .. NVOX/4-1
    const int vb = t * 4;                                   // base voxel index

    // Prefetch streaming inputs ~64KB ahead (gfx1250 global_prefetch_b8;
    // no LOADcnt). Guarded to stay inside the allocations.
    if (vb + 16384 < NVOX) {
        __builtin_prefetch(in + vb + 16384, 0, 3);
        __builtin_prefetch(in + NVOX + vb + 16384, 0, 3);
        __builtin_prefetch(in + 2 * NVOX + vb + 16384, 0, 3);
        __builtin_prefetch(grid + (size_t)(vb + 16384) * 3, 0, 3);
    }

    // Streaming reads: 3 channels of the field (B128 each) ...
    const float4 wx4 = reinterpret_cast<const float4*>(in)[t];
    const float4 wy4 = reinterpret_cast<const float4*>(in + NVOX)[t];
    const float4 wz4 = reinterpret_cast<const float4*>(in + 2 * NVOX)[t];
    // ... and the interleaved identity grid: 4 voxels * 3 comps = 3x float4.
    const float4 g0 = reinterpret_cast<const float4*>(grid)[t * 3 + 0];
    const float4 g1 = reinterpret_cast<const float4*>(grid)[t * 3 + 1];
    const float4 g2 = reinterpret_cast<const float4*>(grid)[t * 3 + 2];

    const float gx[4] = {g0.x, g0.w, g1.z, g2.y};
    const float gy[4] = {g0.y, g1.x, g1.w, g2.z};
    const float gz[4] = {g0.z, g1.y, g2.x, g2.w};
    const float wxv[4] = {wx4.x, wx4.y, wx4.z, wx4.w};
    const float wyv[4] = {wy4.x, wy4.y, wy4.z, wy4.w};
    const float wzv[4] = {wz4.x, wz4.y, wz4.z, wz4.w};

    float ox[4], oy[4], oz[4];

    #pragma unroll
    for (int j = 0; j < 4; ++j) {
        // Unnormalize sample coords exactly as reference:
        // X = (((g + w) + 1) * W - 1) * 0.5
        const float X = 0.5f * (((gx[j] + wxv[j]) + 1.0f) * (float)Ww - 1.0f);
        const float Y = 0.5f * (((gy[j] + wyv[j]) + 1.0f) * (float)Hh - 1.0f);
        const float Z = 0.5f * (((gz[j] + wzv[j]) + 1.0f) * (float)Dd - 1.0f);

        const float xf = floorf(X), yf = floorf(Y), zf = floorf(Z);
        const int ix = (int)xf, iy = (int)yf, iz = (int)zf;
        const float tx = X - xf, ty = Y - yf, tz = Z - zf;

        // Validity-masked per-axis weights (zeros padding).
        const float ux0 = ((unsigned)ix       < (unsigned)Ww) ? (1.0f - tx) : 0.0f;
        const float ux1 = ((unsigned)(ix + 1) < (unsigned)Ww) ? tx          : 0.0f;
        const float uy0 = ((unsigned)iy       < (unsigned)Hh) ? (1.0f - ty) : 0.0f;
        const float uy1 = ((unsigned)(iy + 1) < (unsigned)Hh) ? ty          : 0.0f;
        const float uz0 = ((unsigned)iz       < (unsigned)Dd) ? (1.0f - tz) : 0.0f;
        const float uz1 = ((unsigned)(iz + 1) < (unsigned)Dd) ? tz          : 0.0f;

        // x handled as a B64 pair at sx = clamp(ix, 0, W-2); boundary cases
        // fold into the pair weights (all mismatched corners have weight 0):
        //   interior: p.x <- ux0, p.y <- ux1
        //   ix == -1: live corner is ch[0]   == p.x -> p.x <- ux1
        //   ix == W-1: live corner is ch[W-1] == p.y -> p.y <- ux0
        const int sx = min(max(ix, 0), Ww - 2);
        const float wax = (ix >= Ww - 1) ? 0.0f : ((ix < 0) ? ux1 : ux0);
        const float wbx = (ix >= Ww - 1) ? ux0  : ((ix < 0) ? 0.0f : ux1);

        const int cy0 = min(max(iy,     0), Hh - 1);
        const int cy1 = min(max(iy + 1, 0), Hh - 1);
        const int cz0 = min(max(iz,     0), Dd - 1);
        const int cz1 = min(max(iz + 1, 0), Dd - 1);

        // Four (y,z) row bases, each fetched as one x-pair.
        const int r00 = (cz0 * Hh + cy0) * Ww + sx;
        const int r01 = (cz0 * Hh + cy1) * Ww + sx;
        const int r10 = (cz1 * Hh + cy0) * Ww + sx;
        const int r11 = (cz1 * Hh + cy1) * Ww + sx;

        const float wyz00 = uy0 * uz0, wyz01 = uy1 * uz0;
        const float wyz10 = uy0 * uz1, wyz11 = uy1 * uz1;
        const float wA00 = wax * wyz00, wB00 = wbx * wyz00;
        const float wA01 = wax * wyz01, wB01 = wbx * wyz01;
        const float wA10 = wax * wyz10, wB10 = wbx * wyz10;
        const float wA11 = wax * wyz11, wB11 = wbx * wyz11;

        float accs[3];
        #pragma unroll
        for (int c = 0; c < 3; ++c) {
            const float* ch = in + c * NVOX;   // gathers; L2-resident
            const fpair p00 = *reinterpret_cast<const fpair*>(ch + r00);
            const fpair p01 = *reinterpret_cast<const fpair*>(ch + r01);
            const fpair p10 = *reinterpret_cast<const fpair*>(ch + r10);
            const fpair p11 = *reinterpret_cast<const fpair*>(ch + r11);
            float acc;
            acc = p00.x * wA00;
            acc = fmaf(p00.y, wB00, acc);
            acc = fmaf(p01.x, wA01, acc);
            acc = fmaf(p01.y, wB01, acc);
            acc = fmaf(p10.x, wA10, acc);
            acc = fmaf(p10.y, wB10, acc);
            acc = fmaf(p11.x, wA11, acc);
            acc = fmaf(p11.y, wB11, acc);
            accs[c] = acc;
        }

        ox[j] = wxv[j] + accs[0];
        oy[j] = wyv[j] + accs[1];
        oz[j] = wzv[j] + accs[2];
    }

    reinterpret_cast<float4*>(out)[t]            = make_float4(ox[0], ox[1], ox[2], ox[3]);
    reinterpret_cast<float4*>(out + NVOX)[t]     = make_float4(oy[0], oy[1], oy[2], oy[3]);
    reinterpret_cast<float4*>(out + 2 * NVOX)[t] = make_float4(oz[0], oz[1], oz[2], oz[3]);
}

// -------------------------------------------------------------- launch ------
extern "C" void kernel_launch(void* const* d_in, const int* in_sizes, int n_in,
                              void* d_out, int out_size, void* d_ws, size_t ws_size,
                              hipStream_t stream) {
    const float* grid = (const float*)d_in[0];   // identity_grid [1,D,H,W,3]
    const float* v    = (const float*)d_in[1];   // velocity      [1,3,D,H,W]
    float* out = (float*)d_out;                  // 3*NVOX floats
    float* ws  = (float*)d_ws;                   // >= 3*NVOX floats scratch

    // warp0 = v / 2^12  -> d_out (iteration 1 reads it)
    {
        const int nthreads = (3 * NVOX) / 4;     // 3,072,000 (exact)
        svf_scale_kernel<<<nthreads / 256, 256, 0, stream>>>(v, out);
    }

    // 12 ping-pong iterations; even count => final result lands in d_out.
    const int nthreads = NVOX / 4;               // 1,024,000 (exact)
    for (int k = 1; k <= NO_STEPS; ++k) {
        const float* src = (k & 1) ? out : ws;
        float*       dst = (k & 1) ? ws  : out;
        svf_iter_kernel<<<nthreads / 256, 256, 0, stream>>>(src, grid, dst);
    }
}